// VectorQuantizer_51857435131966
// MI455X (gfx1250) — compile-verified
//
#include <hip/hip_runtime.h>
#include <cfloat>
#include <stdint.h>

typedef float v2f __attribute__((ext_vector_type(2)));
typedef float v8f __attribute__((ext_vector_type(8)));

#define BB 32
#define DD 256
#define HWD 1024           // H*W
#define NN 32768           // B*H*W points
#define KK 1024            // codebook size
#define ZQ_ELEMS 8388608   // B*D*H*W
#define TSTRIDE 260        // padded LDS row stride (floats) -> conflict-free b64 reads

// ---------------------------------------------------------------- zero loss
__global__ __launch_bounds__(64) void vq_zero_kernel(float* acc) {
    if (threadIdx.x == 0) *acc = 0.0f;
}

// ---------------------------------------------------------------- ||e||^2 per code
__global__ __launch_bounds__(256) void vq_enorm_kernel(const float* __restrict__ emb,
                                                       float* __restrict__ eNorm) {
    int wave = (blockIdx.x * 256 + threadIdx.x) >> 5;   // 0..255 (32 blocks)
    int lane = threadIdx.x & 31;
    for (int it = 0; it < 4; ++it) {
        int k = wave * 4 + it;                          // 0..1023
        const float* row = emb + (size_t)k * DD;
        float s = 0.0f;
        for (int d = lane; d < DD; d += 32) { float v = row[d]; s += v * v; }
        for (int off = 16; off > 0; off >>= 1) s += __shfl_xor(s, off, 32);
        if (lane == 0) eNorm[k] = s;
    }
}

// Async-DMA one 16-code x 256-d tile (16 KB) into a padded LDS buffer.
// Each of the 256 threads moves 4 float4 chunks; per wave that is 4
// global_load_async_to_lds_b128 issues tracked by ASYNCcnt (no VGPR round trip).
__device__ __forceinline__ void issue_tile_async(const float* __restrict__ emb,
                                                 int ct, float* dstBase, int tid) {
#pragma unroll
    for (int i = 0; i < 4; ++i) {
        int e4   = tid + i * 256;          // float4 index within tile
        int code = e4 >> 6;                // 64 float4 per code row
        int d4   = e4 & 63;
        uint32_t lds  = (uint32_t)(uintptr_t)(dstBase + code * TSTRIDE + d4 * 4);
        uint32_t goff = (uint32_t)(ct * 16384 + e4 * 16);   // byte offset from emb base
        asm volatile("global_load_async_to_lds_b128 %0, %1, %2"
                     :: "v"(lds), "v"(goff), "s"(emb)
                     : "memory");
    }
}

// ---------------------------------------------------------------- fused GEMM + argmin
// Orientation: A = emb tile (16 codes x 4 d), B = z tile (4 d x 16 points).
// C layout (f32 16x16): lane = point (cols), VGPR i = code row i (+8 for lanes 16-31),
// so argmin over codes is in-lane over 8 VGPRs, plus one xor-16 shuffle at the end.
// Emb tiles are double-buffered in LDS and prefetched with async global->LDS DMA.
__global__ __launch_bounds__(256) void vq_argmin_kernel(const float* __restrict__ z,
                                                        const float* __restrict__ emb,
                                                        const float* __restrict__ eNorm,
                                                        int* __restrict__ idxOut) {
    __shared__ float embS[2][16 * TSTRIDE];  // two 16.25 KB tile buffers
    __shared__ float enAll[KK];              // full ||e||^2 table (4 KB)

    const int lane = threadIdx.x & 31;
    const int wave = threadIdx.x >> 5;         // 8 waves / block
    const int col  = lane & 15;                // point within 16-wide tile
    const int hi   = lane >> 4;                // 0: K/M offset 0, 1: offset 2/8

    // kick off DMA of tile 0 immediately, then stage the norm table
    issue_tile_async(emb, 0, &embS[0][0], threadIdx.x);
    for (int i = threadIdx.x; i < KK; i += 256) enAll[i] = eNorm[i];

    // this wave's 16-point tile; 16 consecutive n never cross a batch image (1024 % 16 == 0)
    const int n  = blockIdx.x * 128 + wave * 16 + col;
    const int b  = n >> 10;
    const int hw = n & 1023;
    const float* zb = z + (size_t)b * (DD * HWD) + hw;

    // Preload B fragments (loop-invariant over code tiles): bfrag[kk] = z[d0], z[d0+1]
    // with d0 = 4*kk + 2*hi. Lanes 0-15 read 16 consecutive hw -> coalesced 64B segments.
    v2f bfrag[64];
#pragma unroll
    for (int kk = 0; kk < 64; ++kk) {
        int d0 = kk * 4 + hi * 2;
        bfrag[kk].x = zb[(size_t)d0 * HWD];
        bfrag[kk].y = zb[(size_t)(d0 + 1) * HWD];
    }

    float best = FLT_MAX;
    int bestIdx = 0;

    for (int ct = 0; ct < 64; ++ct) {          // 64 tiles of 16 codes
        if (ct < 63) {
            // prefetch next tile into the other buffer (its last reader finished
            // at the barrier that ended iteration ct-1), then wait only for the
            // 4 older ops belonging to the current tile
            issue_tile_async(emb, ct + 1, &embS[(ct + 1) & 1][0], threadIdx.x);
            asm volatile("s_wait_asynccnt 4" ::: "memory");
        } else {
            asm volatile("s_wait_asynccnt 0" ::: "memory");
        }
        __syncthreads();                       // current tile visible to all waves

        const float* aB = &embS[ct & 1][col * TSTRIDE + hi * 2];
        v8f acc = {0.f, 0.f, 0.f, 0.f, 0.f, 0.f, 0.f, 0.f};
#pragma unroll
        for (int kk = 0; kk < 64; ++kk) {
            v2f a = *(const v2f*)(aB + kk * 4);      // ds_load_2addr_b64, conflict-free
            acc = __builtin_amdgcn_wmma_f32_16x16x4_f32(
                false, a, false, bfrag[kk], (short)0, acc, false, false);
        }

        // distance proxy = ||e||^2 - 2*(z.e); strict < keeps first index (codes ascend in-lane)
#pragma unroll
        for (int i = 0; i < 8; ++i) {
            int m = i + hi * 8;
            float dist = enAll[ct * 16 + m] - 2.0f * acc[i];
            int ci = ct * 16 + m;
            if (dist < best) { best = dist; bestIdx = ci; }
        }
        __syncthreads();                       // buffer free before it is re-DMA'd
    }

    // merge the two code-halves held by lane L and lane L+16 (same point)
    float ob  = __shfl_xor(best, 16, 32);
    int   obi = __shfl_xor(bestIdx, 16, 32);
    if (ob < best || (ob == best && obi < bestIdx)) { best = ob; bestIdx = obi; }
    if (!hi) idxOut[n] = bestIdx;
}

// ---------------------------------------------------------------- z_q gather + loss sum
__global__ __launch_bounds__(256) void vq_out_kernel(const float* __restrict__ z,
                                                     const float* __restrict__ emb,
                                                     const int* __restrict__ idx,
                                                     float* __restrict__ zq,
                                                     float* __restrict__ lossAcc) {
    __shared__ float red[8];
    const int e0 = (blockIdx.x * 256 + threadIdx.x) * 4;   // element in [B,D,H,W]
    const int b  = e0 >> 18;
    const int d  = (e0 >> 10) & 255;
    const int hw = e0 & 1023;
    const int n0 = (b << 10) + hw;

    float4 zv = *(const float4*)(z + e0);
    int i0 = idx[n0], i1 = idx[n0 + 1], i2 = idx[n0 + 2], i3 = idx[n0 + 3];
    float4 qv;
    qv.x = emb[(size_t)i0 * DD + d];
    qv.y = emb[(size_t)i1 * DD + d];
    qv.z = emb[(size_t)i2 * DD + d];
    qv.w = emb[(size_t)i3 * DD + d];
    *(float4*)(zq + e0) = qv;

    float dx = qv.x - zv.x, dy = qv.y - zv.y, dz = qv.z - zv.z, dw = qv.w - zv.w;
    float s = dx * dx + dy * dy + dz * dz + dw * dw;
    for (int off = 16; off > 0; off >>= 1) s += __shfl_xor(s, off, 32);
    int lane = threadIdx.x & 31, w = threadIdx.x >> 5;
    if (lane == 0) red[w] = s;
    __syncthreads();
    if (threadIdx.x == 0) {
        float t = 0.0f;
        for (int i = 0; i < 8; ++i) t += red[i];
        atomicAdd(lossAcc, t);
    }
}

// ---------------------------------------------------------------- finalize: loss + indices
__global__ __launch_bounds__(256) void vq_finalize_kernel(const int* __restrict__ idx,
                                                          const float* __restrict__ lossAcc,
                                                          float* __restrict__ out) {
    int n = blockIdx.x * 256 + threadIdx.x;
    if (n < NN) out[ZQ_ELEMS + 1 + n] = (float)idx[n];
    if (blockIdx.x == 0 && threadIdx.x == 0)
        out[ZQ_ELEMS] = 1.25f * (*lossAcc) / (float)ZQ_ELEMS;   // (1 + 0.25) * MSE
}

// ----------------------------------------------------------------
extern "C" void kernel_launch(void* const* d_in, const int* in_sizes, int n_in,
                              void* d_out, int out_size, void* d_ws, size_t ws_size,
                              hipStream_t stream) {
    const float* z   = (const float*)d_in[0];   // [32, 256, 32, 32]
    const float* emb = (const float*)d_in[1];   // [1024, 256]
    float* out = (float*)d_out;                 // zq (8388608) | loss (1) | indices (32768)

    float* lossAcc = (float*)d_ws;                                // 4 B
    float* eNorm   = (float*)((char*)d_ws + 64);                  // 4 KB
    int*   idx     = (int*)((char*)d_ws + 64 + 4096);             // 128 KB

    vq_zero_kernel<<<1, 64, 0, stream>>>(lossAcc);
    vq_enorm_kernel<<<32, 256, 0, stream>>>(emb, eNorm);
    vq_argmin_kernel<<<NN / 128, 256, 0, stream>>>(z, emb, eNorm, idx);
    vq_out_kernel<<<ZQ_ELEMS / 1024, 256, 0, stream>>>(z, emb, idx, out, lossAcc);
    vq_finalize_kernel<<<(NN + 255) / 256, 256, 0, stream>>>(idx, lossAcc, out);
}